// CosSimVQ_79525614452863
// MI455X (gfx1250) — compile-verified
//
#include <hip/hip_runtime.h>
#include <hip/hip_bf16.h>

// CosSimVQ forward for MI455X (gfx1250, wave32).
// Pipeline: row-l2norm(x) -> cb GEMM (WMMA f32) -> row-l2norm(cb)
//        -> fused sim GEMM + argmax (WMMA f32, double-buffered LDS fragments)
//        -> rotation trick + deterministic commit-loss reduction.

#define MROWS 32768   // B*N
#define DIMK  512
#define CBN   8192

typedef __attribute__((ext_vector_type(2))) float v2f;
typedef __attribute__((ext_vector_type(8))) float v8f;

__device__ __forceinline__ float waveReduceSum(float v) {
  v += __shfl_xor(v, 1, 32);
  v += __shfl_xor(v, 2, 32);
  v += __shfl_xor(v, 4, 32);
  v += __shfl_xor(v, 8, 32);
  v += __shfl_xor(v, 16, 32);
  return v;
}

__device__ __forceinline__ v8f wmma_f32_k4(v2f a, v2f b, v8f c) {
  return __builtin_amdgcn_wmma_f32_16x16x4_f32(
      /*neg_a=*/false, a, /*neg_b=*/false, b,
      /*c_mod=*/(short)0, c, /*reuse_a=*/false, /*reuse_b=*/false);
}

// ---------------------------------------------------------------------------
// K1 / K2b: per-row L2 normalize (one wave per 512-elem row). eps = 1e-12.
// ---------------------------------------------------------------------------
__global__ void row_l2norm_kernel(const float* __restrict__ in,
                                  float* __restrict__ out, int nrows) {
  const int row = blockIdx.x * 8 + (threadIdx.x >> 5);
  const int lane = threadIdx.x & 31;
  if (row >= nrows) return;
  const float4* rp = (const float4*)(in + (size_t)row * DIMK);
  float4 v0 = rp[lane +  0];
  float4 v1 = rp[lane + 32];
  float4 v2 = rp[lane + 64];
  float4 v3 = rp[lane + 96];
  float ss = v0.x * v0.x + v0.y * v0.y + v0.z * v0.z + v0.w * v0.w
           + v1.x * v1.x + v1.y * v1.y + v1.z * v1.z + v1.w * v1.w
           + v2.x * v2.x + v2.y * v2.y + v2.z * v2.z + v2.w * v2.w
           + v3.x * v3.x + v3.y * v3.y + v3.z * v3.z + v3.w * v3.w;
  ss = waveReduceSum(ss);
  const float denom = fmaxf(sqrtf(ss), 1e-12f);
  float4* op = (float4*)(out + (size_t)row * DIMK);
  float4 o;
  o.x = v0.x / denom; o.y = v0.y / denom; o.z = v0.z / denom; o.w = v0.w / denom; op[lane +  0] = o;
  o.x = v1.x / denom; o.y = v1.y / denom; o.z = v1.z / denom; o.w = v1.w / denom; op[lane + 32] = o;
  o.x = v2.x / denom; o.y = v2.y / denom; o.z = v2.z / denom; o.w = v2.w / denom; op[lane + 64] = o;
  o.x = v3.x / denom; o.y = v3.y / denom; o.z = v3.z / denom; o.w = v3.w / denom; op[lane + 96] = o;
}

// ---------------------------------------------------------------------------
// K2a: implicit_cb = frozen_codebook @ W^T via V_WMMA_F32_16X16X4_F32.
// Each wave computes a 16(code) x 64(dim) tile; fragments loaded from global
// (minor GEMM, 4.3 GFLOP, W is L2-resident).
// Fragment layout (f32, K=4): lane L<16 -> {M=L, K=k0,k0+1}; lane L+16 ->
// {M=L, K=k0+2,k0+3}. B mirrors A with N across lanes: B[k,n] = W[j0+n, k].
// ---------------------------------------------------------------------------
__global__ void cb_transform_kernel(const float* __restrict__ fc,
                                    const float* __restrict__ W,
                                    float* __restrict__ out) {
  const int w = blockIdx.x * 8 + (threadIdx.x >> 5);
  const int lane = threadIdx.x & 31;
  const int c0 = (w >> 3) << 4;   // 512 code tiles
  const int j0 = (w & 7) << 6;    // 8 tiles of 64 output dims
  const int r = lane & 15;
  const int hf = lane >> 4;       // which K half-pair this lane holds

  const float* aB = fc + (size_t)(c0 + r) * DIMK + hf * 2;
  const float* b0 = W + (size_t)(j0 +  0 + r) * DIMK + hf * 2;
  const float* b1 = W + (size_t)(j0 + 16 + r) * DIMK + hf * 2;
  const float* b2 = W + (size_t)(j0 + 32 + r) * DIMK + hf * 2;
  const float* b3 = W + (size_t)(j0 + 48 + r) * DIMK + hf * 2;

  v8f acc0 = {0.f,0.f,0.f,0.f,0.f,0.f,0.f,0.f};
  v8f acc1 = acc0, acc2 = acc0, acc3 = acc0;

  #pragma unroll 4
  for (int k0 = 0; k0 < DIMK; k0 += 4) {
    v2f a = *(const v2f*)(aB + k0);
    acc0 = wmma_f32_k4(a, *(const v2f*)(b0 + k0), acc0);
    acc1 = wmma_f32_k4(a, *(const v2f*)(b1 + k0), acc1);
    acc2 = wmma_f32_k4(a, *(const v2f*)(b2 + k0), acc2);
    acc3 = wmma_f32_k4(a, *(const v2f*)(b3 + k0), acc3);
  }
  // D layout: VGPR v, lanes 0-15 -> row c0+v, lanes 16-31 -> row c0+8+v.
  #pragma unroll
  for (int v = 0; v < 8; ++v) {
    float* o = out + (size_t)(c0 + hf * 8 + v) * DIMK + j0 + r;
    o[0]  = acc0[v];
    o[16] = acc1[v];
    o[32] = acc2[v];
    o[48] = acc3[v];
  }
}

// ---------------------------------------------------------------------------
// LDS fragment-swizzle: element (r, k) of a 16 x 512 tile lives at float
// offset (k>>3)*128 + (r + ((k>>1)&1)*16)*4 + ((k>>2)&1)*2 + (k&1), so one
// ds_load_b128 at (pair*32 + lane) yields this lane's A/B fragments for two
// consecutive K=4 WMMA steps.
// ---------------------------------------------------------------------------
__device__ __forceinline__ void stage_tile_row_quad(float2* base2, int r,
                                                    int kq, float4 v) {
  const int p = kq >> 1;          // k-step-pair index (k0 = kq*4)
  const int slot = kq & 1;        // which k-step within the pair
  base2[p * 64 + r * 2 + slot]        = make_float2(v.x, v.y);  // lanes 0-15
  base2[p * 64 + (r + 16) * 2 + slot] = make_float2(v.z, v.w);  // lanes 16-31
}

// ---------------------------------------------------------------------------
// K3: fused sim GEMM + argmax, software-pipelined.
// Block = 256 threads (8 waves). x block = 64 rows staged once (128 KB,
// fragment order). Wave pair (2m, 2m+1) shares M-tile m and splits each
// 32-code group into two 16-code WMMA tiles. ldsB holds 4 tiles (128 KB):
// two consumed at iteration u, two filled for u+1 via register prefetch, so
// global latency hides behind the 128-WMMA chain; one barrier per iteration.
// __builtin_prefetch warms the u+2 tile (global_prefetch_b8).
// ---------------------------------------------------------------------------
__global__ __launch_bounds__(256) void sim_argmax_kernel(
    const float* __restrict__ xn, const float* __restrict__ cbn,
    int* __restrict__ idx_out) {
  __shared__ __align__(16) float ldsX[64 * DIMK];      // 128 KB, 4 M-tiles
  __shared__ __align__(16) float ldsB[4 * 16 * DIMK];  // 128 KB, 4 code tiles

  const int tid = threadIdx.x;
  const int lane = tid & 31;
  const int w = tid >> 5;        // wave 0..7
  const int mt = w >> 1;         // M-tile 0..3
  const int half16 = w & 1;      // which 16-code subtile this wave scores
  const int rowBase = blockIdx.x * 64;

  // Stage the 64-row x block, pre-swizzled to WMMA fragment order.
  for (int i = tid; i < 64 * (DIMK / 4); i += 256) {
    const int rl = i >> 7, kq = i & 127;
    float4 v = ((const float4*)(xn + (size_t)(rowBase + rl) * DIMK))[kq];
    stage_tile_row_quad((float2*)(ldsX + (rl >> 4) * (16 * DIMK)), rl & 15,
                        kq, v);
  }
  // Stage code tiles for iteration 0 into buffers 0,1.
  for (int i = tid; i < 32 * (DIMK / 4); i += 256) {
    const int rl = i >> 7, kq = i & 127;
    float4 v = ((const float4*)(cbn + (size_t)rl * DIMK))[kq];
    stage_tile_row_quad((float2*)(ldsB + (rl >> 4) * (16 * DIMK)), rl & 15,
                        kq, v);
  }
  __syncthreads();

  const float4* aT = (const float4*)(ldsX + mt * (16 * DIMK));

  float bestv[8];
  int besti[8];
  #pragma unroll
  for (int j = 0; j < 8; ++j) { bestv[j] = -3.4e38f; besti[j] = 0; }

  for (int u = 0; u < CBN / 32; ++u) {
    const int cur = u & 1;
    const bool haveNext = (u + 1) < (CBN / 32);

    // Register-prefetch next 32-code group (64 KB / 256 threads).
    float4 pre[16];
    if (haveNext) {
      const float* src = cbn + (size_t)(u + 1) * 32 * DIMK;
      #pragma unroll
      for (int t2 = 0; t2 < 16; ++t2) {
        const int q = tid + t2 * 256;
        const int rl = q >> 7, kq = q & 127;
        pre[t2] = ((const float4*)(src + (size_t)rl * DIMK))[kq];
      }
      if (u + 2 < (CBN / 32)) {
        const float* pf = cbn + (size_t)(u + 2) * 32 * DIMK;
        __builtin_prefetch(pf + tid * 32, 0, 1);            // 32 KB
        __builtin_prefetch(pf + 8192 + tid * 32, 0, 1);     // next 32 KB
      }
    }

    // 128-step K loop (64 ds_load_b128 pairs) on the current buffer.
    const float4* bT = (const float4*)(ldsB + (cur * 2 + half16) * (16 * DIMK));
    v8f acc = {0.f,0.f,0.f,0.f,0.f,0.f,0.f,0.f};
    #pragma unroll 8
    for (int p = 0; p < 64; ++p) {
      float4 a4 = aT[p * 32 + lane];
      float4 b4 = bT[p * 32 + lane];
      v2f a0 = {a4.x, a4.y}, a1 = {a4.z, a4.w};
      v2f b0 = {b4.x, b4.y}, b1 = {b4.z, b4.w};
      acc = wmma_f32_k4(a0, b0, acc);
      acc = wmma_f32_k4(a1, b1, acc);
    }

    const int code = u * 32 + half16 * 16 + (lane & 15);
    #pragma unroll
    for (int j = 0; j < 8; ++j)
      if (acc[j] > bestv[j]) { bestv[j] = acc[j]; besti[j] = code; }

    // Scatter prefetched registers into the other buffer pair.
    if (haveNext) {
      const int nb = (cur ^ 1) * 2;
      #pragma unroll
      for (int t2 = 0; t2 < 16; ++t2) {
        const int q = tid + t2 * 256;
        const int rl = q >> 7, kq = q & 127;
        stage_tile_row_quad((float2*)(ldsB + (nb + (rl >> 4)) * (16 * DIMK)),
                            rl & 15, kq, pre[t2]);
      }
    }
    __syncthreads();
  }

  // Reduce across the 16 lanes of each half-wave (rows differ by half).
  #pragma unroll
  for (int m = 1; m <= 8; m <<= 1) {
    #pragma unroll
    for (int j = 0; j < 8; ++j) {
      float ov = __shfl_xor(bestv[j], m, 32);
      int   oi = __shfl_xor(besti[j], m, 32);
      if (ov > bestv[j] || (ov == bestv[j] && oi < besti[j])) {
        bestv[j] = ov; besti[j] = oi;
      }
    }
  }

  // Merge the wave pair through an LDS overlay (lower index wins ties).
  float* mv = ldsB;                 // 128 floats
  int*   mi = (int*)(ldsB + 128);   // 128 ints
  const int rsel = lane & 15;
  const int rowOff = (lane >> 4) * 8;  // lanes 16-31 carry rows +8..+15
  #pragma unroll
  for (int j = 0; j < 8; ++j)
    if (rsel == j) {
      mv[w * 16 + rowOff + j] = bestv[j];
      mi[w * 16 + rowOff + j] = besti[j];
    }
  __syncthreads();
  if ((w & 1) == 0) {
    #pragma unroll
    for (int j = 0; j < 8; ++j)
      if (rsel == j) {
        const float vp = mv[(w + 1) * 16 + rowOff + j];
        const int   ip = mi[(w + 1) * 16 + rowOff + j];
        int ii = besti[j];
        if (vp > bestv[j] || (vp == bestv[j] && ip < ii)) ii = ip;
        idx_out[rowBase + mt * 16 + rowOff + j] = ii;
      }
  }
}

// ---------------------------------------------------------------------------
// K4: gather quantized row, rotation trick, per-block commit-loss partial.
// One wave per row. All scalars derived from 4 wave reductions.
// ---------------------------------------------------------------------------
__global__ void rotate_loss_kernel(const float* __restrict__ xn,
                                   const float* __restrict__ cbn,
                                   const int* __restrict__ idx,
                                   float* __restrict__ out_q,
                                   float* __restrict__ out_idx,
                                   float* __restrict__ partials) {
  __shared__ float red[8];
  const int row = blockIdx.x * 8 + (threadIdx.x >> 5);
  const int lane = threadIdx.x & 31;
  const int id = idx[row];
  const float* s = xn + (size_t)row * DIMK;
  const float* t = cbn + (size_t)id * DIMK;

  float sv[16], tv[16];
  float ss = 0.f, tt = 0.f, st = 0.f, d2 = 0.f;
  #pragma unroll
  for (int i = 0; i < 16; ++i) {
    const float a = s[lane + 32 * i];
    const float b = t[lane + 32 * i];
    sv[i] = a; tv[i] = b;
    ss += a * a; tt += b * b; st += a * b;
    const float d = a - b;
    d2 += d * d;
  }
  ss = waveReduceSum(ss);
  tt = waveReduceSum(tt);
  st = waveReduceSum(st);
  d2 = waveReduceSum(d2);

  const float norm_s = sqrtf(ss), norm_t = sqrtf(tt);
  const float inv_s = 1.0f / fmaxf(norm_s, 1e-6f);
  const float inv_t = 1.0f / fmaxf(norm_t, 1e-6f);
  // ||u + q||^2 = ss/ns^2 + 2*st/(ns*nt) + tt/nt^2
  const float wn2 = ss * inv_s * inv_s + 2.f * st * inv_s * inv_t +
                    tt * inv_t * inv_t;
  const float inv_w = 1.0f / fmaxf(sqrtf(wn2), 1e-12f);
  const float ew = (ss * inv_s + st * inv_t) * inv_w;  // sum(e*w)
  const float eu = ss * inv_s;                         // sum(e*u)
  const float scale = norm_t * inv_s;

  float* o = out_q + (size_t)row * DIMK;
  #pragma unroll
  for (int i = 0; i < 16; ++i) {
    const float wi = (sv[i] * inv_s + tv[i] * inv_t) * inv_w;
    o[lane + 32 * i] =
        (sv[i] - 2.f * ew * wi + 2.f * eu * tv[i] * inv_t) * scale;
  }
  if (lane == 0) {
    out_idx[row] = (float)id;
    red[threadIdx.x >> 5] = d2;
  }
  __syncthreads();
  if (threadIdx.x == 0) {
    float a = 0.f;
    #pragma unroll
    for (int i = 0; i < 8; ++i) a += red[i];
    partials[blockIdx.x] = a;
  }
}

// ---------------------------------------------------------------------------
// K5: deterministic loss finalize. loss = 1.25 * sum(d2) / (M*D).
// ---------------------------------------------------------------------------
__global__ void loss_finalize_kernel(const float* __restrict__ partials, int n,
                                     float* __restrict__ out_loss) {
  __shared__ float red[256];
  float a = 0.f;
  for (int i = threadIdx.x; i < n; i += 256) a += partials[i];
  red[threadIdx.x] = a;
  __syncthreads();
  for (int sft = 128; sft > 0; sft >>= 1) {
    if (threadIdx.x < sft) red[threadIdx.x] += red[threadIdx.x + sft];
    __syncthreads();
  }
  if (threadIdx.x == 0)
    out_loss[0] = red[0] * (1.25f / (float)((size_t)MROWS * DIMK));
}

// ---------------------------------------------------------------------------
extern "C" void kernel_launch(void* const* d_in, const int* in_sizes, int n_in,
                              void* d_out, int out_size, void* d_ws,
                              size_t ws_size, hipStream_t stream) {
  const float* x = (const float*)d_in[0];
  const float* W = (const float*)d_in[1];
  const float* fc = (const float*)d_in[2];

  float* ws = (float*)d_ws;
  float* x_norm = ws;                                    // 32768*512 f32
  float* cb = ws + (size_t)MROWS * DIMK;                 // 8192*512 f32
  int* idx = (int*)(cb + (size_t)CBN * DIMK);            // 32768 i32
  float* partials = (float*)(idx + MROWS);               // 4096 f32

  float* out_q = (float*)d_out;                          // (8,4096,512)
  float* out_idx = out_q + (size_t)MROWS * DIMK;         // (8,4096) as f32
  float* out_loss = out_idx + MROWS;                     // scalar

  hipLaunchKernelGGL(row_l2norm_kernel, dim3(MROWS / 8), dim3(256), 0, stream,
                     x, x_norm, MROWS);
  hipLaunchKernelGGL(cb_transform_kernel, dim3(512), dim3(256), 0, stream,
                     fc, W, cb);
  hipLaunchKernelGGL(row_l2norm_kernel, dim3(CBN / 8), dim3(256), 0, stream,
                     cb, cb, CBN);
  hipLaunchKernelGGL(sim_argmax_kernel, dim3(MROWS / 64), dim3(256), 0, stream,
                     x_norm, cb, idx);
  hipLaunchKernelGGL(rotate_loss_kernel, dim3(MROWS / 8), dim3(256), 0, stream,
                     x_norm, cb, idx, out_q, out_idx, partials);
  hipLaunchKernelGGL(loss_finalize_kernel, dim3(1), dim3(256), 0, stream,
                     partials, MROWS / 8, out_loss);
}